// ForwardKinematics_29463475651255
// MI455X (gfx1250) — compile-verified
//
#include <hip/hip_runtime.h>
#include <hip/hip_bf16.h>
#include <stdint.h>

// Forward kinematics, 524288 frames x 12 joints (two 6-joint serial chains).
// Bandwidth-bound: ~215 MB total traffic -> ~9.2 us floor at 23.3 TB/s.
// CDNA5 paths: TDM tensor_load_to_lds (TENSORcnt) stages each block's
// 256x12 fp32 qpos tile into LDS; s_wait_tensorcnt; non-temporal streaming
// stores for the 191 MB write-once output (protects the 192 MB L2).
// TDM issued FIRST so the DMA overlaps constant staging + root-joint stores.

#define FK_FRAMES 524288
#define NJ 12
#define FPB 256   // frames per block (8 waves of 32)

typedef unsigned int u32x4 __attribute__((ext_vector_type(4)));
typedef int          i32x4 __attribute__((ext_vector_type(4)));
typedef int          i32x8 __attribute__((ext_vector_type(8)));
typedef float        v4f   __attribute__((ext_vector_type(4)));

__global__ __launch_bounds__(FPB)
void fk_kernel(const float* __restrict__ qpos,
               const float* __restrict__ body_pos,
               const float* __restrict__ body_axis,
               const float* __restrict__ body_range,
               float* __restrict__ out)
{
    __shared__ float sq[FPB * NJ];      // 12 KB qpos tile, TDM-filled
    __shared__ float s_pos[NJ * 3];
    __shared__ float s_axis[NJ * 3];
    __shared__ float s_range[NJ * 2];

    const int tid = threadIdx.x;

    // ---- 1) Kick off the TDM DMA immediately (wave 0 only). ----
    // 2D tensor, 4-byte elems, tile 12 x 256 (x = joint contiguous, y = frame),
    // tensor_dim = tile_dim (tile exactly covers it, no OOB), stride0 = 12.
    if (tid < 32) {
        uint64_t gaddr   = (uint64_t)(uintptr_t)(qpos + (size_t)blockIdx.x * FPB * NJ);
        uint32_t lds_off = (uint32_t)(uintptr_t)(&sq[0]); // low 32 bits = LDS offset

        u32x4 g0;
        g0[0] = 1u;                                          // count = 1 valid D#
        g0[1] = lds_off;                                     // lds_addr
        g0[2] = (uint32_t)gaddr;                             // global_addr[31:0]
        g0[3] = ((uint32_t)(gaddr >> 32) & 0x01FFFFFFu)      // global_addr[56:32]
              | (2u << 30);                                  // type = 2 ("image")

        i32x8 g1;
        g1[0] = 0x00020000;        // data_size = 2 (4 bytes); no mask/pad/iterate
        g1[1] = (NJ  << 16);       // tensor_dim0[15:0] = 12
        g1[2] = (FPB << 16);       // tensor_dim0 hi = 0 | tensor_dim1[15:0] = 256
        g1[3] = (NJ  << 16);       // tensor_dim1 hi = 0 | tile_dim0 = 12
        g1[4] = FPB;               // tile_dim1 = 256, tile_dim2 = 0
        g1[5] = NJ;                // tensor_dim0_stride = 12
        g1[6] = 0;
        g1[7] = 0;

        i32x4 g2 = {0, 0, 0, 0};   // 2D tensor: groups 2/3 unused
        i32x4 g3 = {0, 0, 0, 0};

#if defined(__clang_major__) && (__clang_major__ >= 23)
        i32x8 g4 = {0, 0, 0, 0, 0, 0, 0, 0};
        __builtin_amdgcn_tensor_load_to_lds(g0, g1, g2, g3, g4, 0);
#else
        __builtin_amdgcn_tensor_load_to_lds(g0, g1, g2, g3, 0);
#endif
    }

    // ---- 2) Overlap the DMA: stage constants (96 floats) into LDS. ----
    if (tid < NJ * 3)        s_pos[tid]            = body_pos[tid];
    else if (tid < NJ * 6)   s_axis[tid - NJ * 3]  = body_axis[tid - NJ * 3];
    else if (tid < NJ * 8)   s_range[tid - NJ * 6] = body_range[tid - NJ * 6];

    const int f = blockIdx.x * FPB + tid;

    float* __restrict__ out_pos = out;
    float* __restrict__ out_rot = out + (size_t)13 * FK_FRAMES * 3;

    // ---- 3) Overlap the DMA: root joint (index 0) stores need no LDS. ----
    {
        size_t pb = (size_t)f * 3;
        __builtin_nontemporal_store(0.0f, &out_pos[pb + 0]);
        __builtin_nontemporal_store(0.0f, &out_pos[pb + 1]);
        __builtin_nontemporal_store(0.0f, &out_pos[pb + 2]);
        v4f rq = {0.0f, 0.0f, 0.0f, 1.0f};
        __builtin_nontemporal_store(rq, (v4f*)&out_rot[(size_t)f * 4]);
    }

    // ---- 4) Wave 0 drains TENSORcnt; everyone syncs on the tile. ----
    if (tid < 32) {
        __builtin_amdgcn_s_wait_tensorcnt(0);
    }
    __syncthreads();

    float q[NJ];
#pragma unroll
    for (int j = 0; j < NJ; ++j) q[j] = sq[tid * NJ + j];   // 3x ds_load_b128

    // Running parent state (both chains start at root).
    float px = 0.f, py = 0.f, pz = 0.f;
    float rx = 0.f, ry = 0.f, rz = 0.f, rw = 1.f;

#pragma unroll
    for (int j = 0; j < NJ; ++j) {
        if (j == 6) { px = py = pz = 0.f; rx = ry = rz = 0.f; rw = 1.f; }

        // dof = clip(qpos, lo, hi); joint quat from angle-axis.
        float d = fminf(fmaxf(q[j], s_range[2 * j]), s_range[2 * j + 1]);
        float s, c;
        __sincosf(0.5f * d, &s, &c);
        float ax = s_axis[3 * j], ay = s_axis[3 * j + 1], az = s_axis[3 * j + 2];
        float jx = s * ax, jy = s * ay, jz = s * az, jw = c;

        // pos[j] = parent_pos + quat_rotate(parent_rot, body_pos[j])
        float vx = s_pos[3 * j], vy = s_pos[3 * j + 1], vz = s_pos[3 * j + 2];
        float ww = 2.f * rw * rw - 1.f;
        float cx = ry * vz - rz * vy;
        float cy = rz * vx - rx * vz;
        float cz = rx * vy - ry * vx;
        float dt = rx * vx + ry * vy + rz * vz;
        px += vx * ww + 2.f * rw * cx + 2.f * rx * dt;
        py += vy * ww + 2.f * rw * cy + 2.f * ry * dt;
        pz += vz * ww + 2.f * rw * cz + 2.f * rz * dt;

        // rot[j] = quat_mul(parent_rot, joint_rot)
        float ox = rw * jx + rx * jw + ry * jz - rz * jy;
        float oy = rw * jy - rx * jz + ry * jw + rz * jx;
        float oz = rw * jz + rx * jy - ry * jx + rz * jw;
        float ow = rw * jw - rx * jx - ry * jy - rz * jz;
        rx = ox; ry = oy; rz = oz; rw = ow;

        // Streaming stores: [13][FRAMES][3] and [13][FRAMES][4] layouts.
        size_t pb = ((size_t)(j + 1) * FK_FRAMES + f) * 3;
        __builtin_nontemporal_store(px, &out_pos[pb + 0]);   // merges to b96
        __builtin_nontemporal_store(py, &out_pos[pb + 1]);
        __builtin_nontemporal_store(pz, &out_pos[pb + 2]);
        v4f r4 = {rx, ry, rz, rw};
        __builtin_nontemporal_store(r4,
            (v4f*)&out_rot[((size_t)(j + 1) * FK_FRAMES + f) * 4]);
    }
}

extern "C" void kernel_launch(void* const* d_in, const int* in_sizes, int n_in,
                              void* d_out, int out_size, void* d_ws, size_t ws_size,
                              hipStream_t stream) {
    (void)in_sizes; (void)n_in; (void)out_size; (void)d_ws; (void)ws_size;
    const float* qpos       = (const float*)d_in[0];
    const float* body_pos   = (const float*)d_in[1];
    const float* body_axis  = (const float*)d_in[2];
    const float* body_range = (const float*)d_in[3];
    float* out = (float*)d_out;

    dim3 grid(FK_FRAMES / FPB), block(FPB);
    hipLaunchKernelGGL(fk_kernel, grid, block, 0, stream,
                       qpos, body_pos, body_axis, body_range, out);
}